// Signature_82119774699924
// MI455X (gfx1250) — compile-verified
//
#include <hip/hip_runtime.h>

typedef __attribute__((ext_vector_type(2))) float v2f;
typedef __attribute__((ext_vector_type(8))) float v8f;

#if defined(__has_builtin)
#if __has_builtin(__builtin_amdgcn_s_wait_asynccnt)
#define HAVE_WAIT_ASYNC 1
#endif
#endif

#define NPATH 256
#define LPATH 256
#define CH    8
#define D3    512          // 8^3 rows of level-4 viewed as [512 x 8]
#define OUT_STRIDE 4680    // 8 + 64 + 512 + 4096
#define OFF2  8
#define OFF3  72
#define OFF4  584

__global__ __launch_bounds__(128)
void sig_depth4_kernel(const float* __restrict__ path, float* __restrict__ out) {
    const int p   = blockIdx.x;
    const int tid = threadIdx.x;

    __shared__ float pathS[LPATH * CH];  // 8 KB: whole path staged once
    __shared__ float Pbuf[2][4][D3];     // 16 KB: P_t, double-buffered per chunk parity
    __shared__ float vbuf[2][4][CH];     // increments, double-buffered per chunk parity
    __shared__ float s1[2][CH];          // level-1 prefix, ping-pong per step parity
    __shared__ float s2[2][64];          // level-2 prefix, ping-pong per step parity

    // ---- stage path[p] into LDS via async global->LDS (ASYNCcnt, no VGPR round-trip) ----
    const float4* gp = (const float4*)(path + (size_t)p * LPATH * CH);
    float4* sp = (float4*)pathS;
#pragma unroll
    for (int it = 0; it < 4; ++it) {
        const float4*  ga      = gp + tid + 128 * it;
        unsigned       lds_off = (unsigned)(size_t)(sp + tid + 128 * it);
        asm volatile("global_load_async_to_lds_b128 %0, %1, off"
                     :
                     : "v"(lds_off), "v"(ga)
                     : "memory");
    }
#if defined(HAVE_WAIT_ASYNC)
    __builtin_amdgcn_s_wait_asynccnt(0);
#else
    asm volatile("s_wait_asynccnt 0" ::: "memory");
#endif

    if (tid < 8)  s1[0][tid] = 0.f;
    if (tid < 64) s2[0][tid] = 0.f;

    // level-3 prefix: thread tid owns rows tid*4 .. tid*4+3 (ijk index)
    float s3r[4] = {0.f, 0.f, 0.f, 0.f};

    // level-4 accumulators: wave w owns row-tiles w*8 .. w*8+7 ([16 x 16] each)
    const v8f vzero = {0.f, 0.f, 0.f, 0.f, 0.f, 0.f, 0.f, 0.f};
    v8f acc[8];
#pragma unroll
    for (int r = 0; r < 8; ++r) acc[r] = vzero;

    const int ij = tid >> 1;        // shared (i,j) pair for this thread's 4 rows
    const int ii = ij >> 3;
    const int jj = ij & 7;
    const int k0 = (tid & 1) * 4;   // k range base

    const int lane  = tid & 31;
    const int wv    = tid >> 5;
    const int colB  = lane & 15;    // N index for B / D fragments
    const int halfB = lane >> 4;    // lane-half selects K pair / M half

    __syncthreads();  // staged path + cleared prefixes visible

    // 64 chunks x 4 steps = 256 steps; step 255 is a zero-increment pad.
    // One barrier per step (ping-pong prefixes) + one per chunk (pre-WMMA).
    for (int chunk = 0; chunk < 64; ++chunk) {
        const int cb = chunk & 1;
#pragma unroll
        for (int sub = 0; sub < 4; ++sub) {
            const int t   = chunk * 4 + sub;
            const int cur = t & 1;
            const int nxt = cur ^ 1;
            if (tid < 8) {
                float v = 0.f;
                if (t < LPATH - 1)
                    v = pathS[(t + 1) * CH + tid] - pathS[t * CH + tid];
                vbuf[cb][sub][tid] = v;
            }
            __syncthreads(); // vbuf[cb][sub] + s1/s2[cur] (written at t-1) visible

            const float vi   = vbuf[cb][sub][ii];
            const float vj   = vbuf[cb][sub][jj];
            const float s1i  = s1[cur][ii];
            const float s2ij = s2[cur][ij];
            // s3 += M (x) v ;  P = s3_prev + Q (x) v   (grouped by last index)
            const float m = s2ij        + 0.5f         * s1i * vj + (1.f / 6.f)  * vi * vj;
            const float q = 0.5f * s2ij + (1.f / 6.f)  * s1i * vj + (1.f / 24.f) * vi * vj;
#pragma unroll
            for (int kk = 0; kk < 4; ++kk) {
                const float vk = vbuf[cb][sub][k0 + kk];
                Pbuf[cb][sub][tid * 4 + kk] = s3r[kk] + q * vk; // level-4 GEMM A rows
                s3r[kk] += m * vk;                              // level-3 prefix update
            }
            // level-1/2 prefix updates into the opposite parity (no extra barrier)
            if (tid < 64)
                s2[nxt][tid] = s2[cur][tid]
                             + s1[cur][tid >> 3] * vbuf[cb][sub][tid & 7]
                             + 0.5f * vbuf[cb][sub][tid >> 3] * vbuf[cb][sub][tid & 7];
            if (tid < 8)
                s1[nxt][tid] = s1[cur][tid] + vbuf[cb][sub][tid];
        }
        __syncthreads(); // all 4 P_t / v_t slices of this chunk visible

        // ---- level-4: sig4 += [512 x 4] @ [4 x 8] via v_wmma_f32_16x16x4_f32 ----
        // B (4x16): lanes 0-15 -> N, VGPR0/1 = K0/K1; lanes 16-31 -> K2/K3. Cols 8-15 zero.
        v2f bfrag;
        bfrag.x = (colB < 8) ? vbuf[cb][2 * halfB + 0][colB] : 0.f;
        bfrag.y = (colB < 8) ? vbuf[cb][2 * halfB + 1][colB] : 0.f;

#pragma unroll
        for (int rr = 0; rr < 8; ++rr) {
            const int row = (wv * 8 + rr) * 16 + (lane & 15);
            // A (16x4): lanes 0-15 rows M, VGPR0/1 = K0/K1; lanes 16-31 = K2/K3
            v2f afrag;
            afrag.x = Pbuf[cb][2 * halfB + 0][row];
            afrag.y = Pbuf[cb][2 * halfB + 1][row];
            acc[rr] = __builtin_amdgcn_wmma_f32_16x16x4_f32(
                /*neg_a=*/false, afrag, /*neg_b=*/false, bfrag,
                /*c_mod=*/(short)0, acc[rr], /*reuse_a=*/false, /*reuse_b=*/false);
        }
        // no end-of-chunk barrier needed: next chunk writes the other Pbuf/vbuf parity,
        // and the WMMA fragment loads are separated from that chunk's writes by its
        // own per-step barriers.
    }

    __syncthreads();

    // ---- write out: [s1 | s2 | s3 | s4] per path ----
    // After 256 steps the live prefix parity is (256 & 1) == 0.
    const size_t base = (size_t)p * OUT_STRIDE;
    if (tid < 8)  out[base + tid]        = s1[0][tid];
    if (tid < 64) out[base + OFF2 + tid] = s2[0][tid];
#pragma unroll
    for (int kk = 0; kk < 4; ++kk)
        out[base + OFF3 + tid * 4 + kk] = s3r[kk];

    // D (16x16 f32): lane 0-15 -> N=lane, VGPR j -> M=j; lanes 16-31 -> M=8+j
    if (colB < 8) {
#pragma unroll
        for (int rr = 0; rr < 8; ++rr) {
#pragma unroll
            for (int jr = 0; jr < 8; ++jr) {
                const int row = (wv * 8 + rr) * 16 + halfB * 8 + jr;
                out[base + OFF4 + row * 8 + colB] = acc[rr][jr];
            }
        }
    }
}

extern "C" void kernel_launch(void* const* d_in, const int* in_sizes, int n_in,
                              void* d_out, int out_size, void* d_ws, size_t ws_size,
                              hipStream_t stream) {
    (void)in_sizes; (void)n_in; (void)d_ws; (void)ws_size; (void)out_size;
    const float* path = (const float*)d_in[0]; // [256, 256, 8] f32
    float* out = (float*)d_out;                // [256, 4680] f32
    sig_depth4_kernel<<<dim3(NPATH), dim3(128), 0, stream>>>(path, out);
}